// meta_optimizer_7507602833727
// MI455X (gfx1250) — compile-verified
//
#include <hip/hip_runtime.h>
#include <math.h>
#include <stdint.h>

typedef __attribute__((ext_vector_type(2))) float v2f;
typedef __attribute__((ext_vector_type(8))) float v8f;
typedef __attribute__((ext_vector_type(4))) int   v4i;

#define WD_C    0.0005f
#define SCALE_C 0.1f
#define KD      4608      // k = 3*3*512
#define N0      512
#define HD      20

#if defined(__HIP_DEVICE_COMPILE__) && __has_builtin(__builtin_amdgcn_global_load_async_to_lds_b128)
#define HAVE_ASYNC_LDS 1
#define AS1 __attribute__((address_space(1)))
#define AS3 __attribute__((address_space(3)))
__device__ __forceinline__ AS1 v4i* to_glob_v4(const void* p) {
  return (AS1 v4i*)(uintptr_t)p;           // drops const; addrspace(1) = global
}
__device__ __forceinline__ AS3 v4i* to_lds_v4(void* p) {
  return (AS3 v4i*)(uint32_t)(uintptr_t)p; // addrspace(3) = LDS
}
#else
#define HAVE_ASYNC_LDS 0
#endif

// ============================================================
// 1) new_e = e - e_lr * (e_grad + WD*e)    (float4 vectorized)
// ============================================================
__global__ void k_e_update(const float* __restrict__ e, const float* __restrict__ g,
                           const float* __restrict__ lr, float* __restrict__ out, int n4) {
  int i = blockIdx.x * blockDim.x + threadIdx.x;
  if (i >= n4) return;
  float l = *lr;
  float4 ev = ((const float4*)e)[i];
  float4 gv = ((const float4*)g)[i];
  float4 o;
  o.x = ev.x - l * (gv.x + WD_C * ev.x);
  o.y = ev.y - l * (gv.y + WD_C * ev.y);
  o.z = ev.z - l * (gv.z + WD_C * ev.z);
  o.w = ev.w - l * (gv.w + WD_C * ev.w);
  ((float4*)out)[i] = o;
}

// ============================================================
// 2) C(MxN) = alpha * A(MxK) * B(NxK)^T     fp32 WMMA 16x16x4
//    simple: one 16x16 tile per wave (used for 512x512 outputs,
//    where K is deep and wave-parallelism matters most)
// ============================================================
__global__ void k_gemm_abt(const float* __restrict__ A, const float* __restrict__ B,
                           float* __restrict__ C, int M, int N, int K,
                           float alpha_h, const float* __restrict__ alpha_p) {
  int wave = (blockIdx.x * blockDim.x + threadIdx.x) >> 5;
  int lane = threadIdx.x & 31;
  int tnc = N >> 4;
  int tm = wave / tnc;
  int tn = wave - tm * tnc;
  if (tm >= (M >> 4)) return;          // wave-uniform; EXEC stays all-ones
  int l15 = lane & 15, lh = lane >> 4;
  const float* ap = A + (size_t)(tm * 16 + l15) * K + 2 * lh;
  const float* bp = B + (size_t)(tn * 16 + l15) * K + 2 * lh;
  v8f acc0 = {}, acc1 = {};
  for (int k = 0; k < K; k += 8) {
    v2f a0 = *(const v2f*)(ap + k);
    v2f b0 = *(const v2f*)(bp + k);
    v2f a1 = *(const v2f*)(ap + k + 4);
    v2f b1 = *(const v2f*)(bp + k + 4);
    acc0 = __builtin_amdgcn_wmma_f32_16x16x4_f32(false, a0, false, b0, (short)0, acc0, false, false);
    acc1 = __builtin_amdgcn_wmma_f32_16x16x4_f32(false, a1, false, b1, (short)0, acc1, false, false);
  }
  v8f acc = acc0 + acc1;
  float alpha = alpha_h * (alpha_p ? *alpha_p : 1.0f);
  float* cp = C + (size_t)(tm * 16 + 8 * lh) * N + tn * 16 + l15;
#pragma unroll
  for (int r = 0; r < 8; ++r) cp[(size_t)r * N] = alpha * acc[r];
}

// ============================================================
// 2b) cooperative A*B^T GEMM for the big (512x4608x4608) case:
//     block = 8 waves = 128(M) x 64(N) tile; B tile staged in
//     LDS (async-to-LDS when available) and shared by all waves;
//     each wave: 32x32 out = 4 x v8f accumulators.
// ============================================================
#define KC   64
#define BPAD 4    // row stride 68 dwords: 16B-aligned rows + conflict-free ds reads

__global__ void __launch_bounds__(256)
k_gemm_abt_coop(const float* __restrict__ A, const float* __restrict__ B,
                float* __restrict__ C, int M, int N, int K, float alpha) {
  __shared__ float smemB[64][KC + BPAD];
  int tid  = threadIdx.x;
  int wave = tid >> 5, lane = tid & 31;
  int l15 = lane & 15, lh = lane >> 4;
  int wm = wave >> 1;                     // 0..3 (M dir)
  int wn = wave & 1;                      // 0..1 (N dir)
  int mb     = blockIdx.y * 128 + wm * 32;
  int nb_blk = blockIdx.x * 64;
  int n_loc  = wn * 32;
  const float* ap0 = A + (size_t)(mb + l15) * K + 2 * lh;
  const float* ap1 = ap0 + (size_t)16 * K;
  // staging map: thread t -> B row (t>>2), 16 consecutive floats at col (t&3)*16
  int sr = tid >> 2, sc = (tid & 3) * 16;
  const float* gstage = B + (size_t)(nb_blk + sr) * K + sc;
  float* lstage = &smemB[sr][sc];

  v8f acc00 = {}, acc01 = {}, acc10 = {}, acc11 = {};
  for (int k0 = 0; k0 < K; k0 += KC) {
    __syncthreads();   // previous chunk's readers done before restage
#if HAVE_ASYNC_LDS
#pragma unroll
    for (int q = 0; q < 4; ++q)
      __builtin_amdgcn_global_load_async_to_lds_b128(
          to_glob_v4(gstage + k0 + q * 4), to_lds_v4(lstage + q * 4), 0, 0);
#if __has_builtin(__builtin_amdgcn_s_wait_asynccnt)
    __builtin_amdgcn_s_wait_asynccnt(0);
#else
    asm volatile("s_wait_asynccnt 0" ::: "memory");
#endif
#else
#pragma unroll
    for (int q = 0; q < 4; ++q) {
      float4 v = *(const float4*)(gstage + k0 + q * 4);
      *(float4*)(lstage + q * 4) = v;
    }
#endif
    __syncthreads();
#pragma unroll
    for (int kk = 0; kk < KC; kk += 4) {
      v2f a0 = *(const v2f*)(ap0 + k0 + kk);
      v2f a1 = *(const v2f*)(ap1 + k0 + kk);
      v2f b0 = *(const v2f*)(&smemB[n_loc + l15][kk + 2 * lh]);
      v2f b1 = *(const v2f*)(&smemB[n_loc + 16 + l15][kk + 2 * lh]);
      acc00 = __builtin_amdgcn_wmma_f32_16x16x4_f32(false, a0, false, b0, (short)0, acc00, false, false);
      acc01 = __builtin_amdgcn_wmma_f32_16x16x4_f32(false, a0, false, b1, (short)0, acc01, false, false);
      acc10 = __builtin_amdgcn_wmma_f32_16x16x4_f32(false, a1, false, b0, (short)0, acc10, false, false);
      acc11 = __builtin_amdgcn_wmma_f32_16x16x4_f32(false, a1, false, b1, (short)0, acc11, false, false);
    }
  }
  float* cp = C + (size_t)(mb + 8 * lh) * N + nb_blk + n_loc + l15;
#pragma unroll
  for (int r = 0; r < 8; ++r) {
    cp[(size_t)r * N]              = alpha * acc00[r];
    cp[(size_t)r * N + 16]         = alpha * acc01[r];
    cp[(size_t)(r + 16) * N]       = alpha * acc10[r];
    cp[(size_t)(r + 16) * N + 16]  = alpha * acc11[r];
  }
}

// ============================================================
// 3) C(MxN) = alpha*A(MxK)*B(KxN) + c0*Add0 + c1*Add1  (K=512)
// ============================================================
__global__ void k_gemm_ab(const float* __restrict__ A, const float* __restrict__ B,
                          float* __restrict__ C,
                          const float* __restrict__ add0, float c0h, const float* __restrict__ c0p,
                          const float* __restrict__ add1, float c1h,
                          int M, int N, int K, float alpha) {
  int wave = (blockIdx.x * blockDim.x + threadIdx.x) >> 5;
  int lane = threadIdx.x & 31;
  int tnc = N >> 4;
  int tm = wave / tnc;
  int tn = wave - tm * tnc;
  if (tm >= (M >> 4)) return;
  int l15 = lane & 15, lh = lane >> 4;
  const float* ap = A + (size_t)(tm * 16 + l15) * K + 2 * lh;
  int col = tn * 16 + l15;
  v8f acc0 = {}, acc1 = {};
  for (int k = 0; k < K; k += 8) {
    v2f a0 = *(const v2f*)(ap + k);
    v2f a1 = *(const v2f*)(ap + k + 4);
    int kb = k + 2 * lh;
    v2f b0 = { B[(size_t)(kb)*N + col],       B[(size_t)(kb + 1) * N + col] };
    v2f b1 = { B[(size_t)(kb + 4) * N + col], B[(size_t)(kb + 5) * N + col] };
    acc0 = __builtin_amdgcn_wmma_f32_16x16x4_f32(false, a0, false, b0, (short)0, acc0, false, false);
    acc1 = __builtin_amdgcn_wmma_f32_16x16x4_f32(false, a1, false, b1, (short)0, acc1, false, false);
  }
  v8f acc = acc0 + acc1;
  float c0 = c0h * (c0p ? *c0p : 1.0f);
#pragma unroll
  for (int r = 0; r < 8; ++r) {
    size_t idx = (size_t)(tm * 16 + 8 * lh + r) * N + col;
    float v = alpha * acc[r];
    if (add0) v += c0 * add0[idx];
    if (add1) v += c1h * add1[idx];
    C[idx] = v;
  }
}

// ============================================================
// 4) out = 0.5*(in + in^T), 512x512
// ============================================================
__global__ void k_symmetrize(const float* __restrict__ in, float* __restrict__ out) {
  int idx = blockIdx.x * blockDim.x + threadIdx.x;
  if (idx >= N0 * N0) return;
  int i = idx / N0, j = idx - i * N0;
  out[idx] = 0.5f * (in[i * N0 + j] + in[j * N0 + i]);
}

// ============================================================
// 5) x[i] = clip(log(|sum_a Gt[a][i]^2|)/P, -1, 1)
// ============================================================
__global__ void k_coldiag(const float* __restrict__ Gt, float* __restrict__ X) {
  int i = blockIdx.x * blockDim.x + threadIdx.x;
  if (i >= KD) return;
  float s = 0.f;
  for (int a = 0; a < N0; ++a) { float v = Gt[(size_t)a * KD + i]; s += v * v; }
  float x = logf(fabsf(s)) * (1.0f / 10.0f);
  X[i] = fminf(1.0f, fmaxf(-1.0f, x));
}

// ============================================================
// 6) 2-layer LSTM cell, batch=KD, H=20 (one thread per row)
// ============================================================
__device__ __forceinline__ float sigf(float x) { return 1.0f / (1.0f + expf(-x)); }

__global__ void k_lstm(const float* __restrict__ X,
                       const float* __restrict__ h0all, const float* __restrict__ c0all,
                       const float* __restrict__ Wih0, const float* __restrict__ Whh0,
                       const float* __restrict__ bih0, const float* __restrict__ bhh0,
                       const float* __restrict__ Wih1, const float* __restrict__ Whh1,
                       const float* __restrict__ bih1, const float* __restrict__ bhh1,
                       const float* __restrict__ Wl, const float* __restrict__ bl,
                       float* __restrict__ hn_out, float* __restrict__ cn_out,
                       float* __restrict__ lpre) {
  int i = blockIdx.x * blockDim.x + threadIdx.x;
  if (i >= KD) return;
  const int KH = KD * HD;
  float xv = X[i];
  float h0[HD], c0v[HD], h1p[HD], c1p[HD];
  for (int h = 0; h < HD; ++h) {
    h0[h]  = h0all[i * HD + h];
    c0v[h] = c0all[i * HD + h];
    h1p[h] = h0all[KH + i * HD + h];
    c1p[h] = c0all[KH + i * HD + h];
  }
  float hn0[HD];
  for (int h = 0; h < HD; ++h) {
    float gi = Wih0[h] * xv, gf = Wih0[HD + h] * xv;
    float gg = Wih0[2 * HD + h] * xv, go = Wih0[3 * HD + h] * xv;
    for (int p = 0; p < HD; ++p) {
      float hp = h0[p];
      gi += Whh0[(h)*HD + p] * hp;
      gf += Whh0[(HD + h) * HD + p] * hp;
      gg += Whh0[(2 * HD + h) * HD + p] * hp;
      go += Whh0[(3 * HD + h) * HD + p] * hp;
    }
    gi += bih0[h] + bhh0[h];                   gf += bih0[HD + h] + bhh0[HD + h];
    gg += bih0[2 * HD + h] + bhh0[2 * HD + h]; go += bih0[3 * HD + h] + bhh0[3 * HD + h];
    float c = sigf(gf) * c0v[h] + sigf(gi) * tanhf(gg);
    float hh = sigf(go) * tanhf(c);
    hn0[h] = hh;
    hn_out[i * HD + h] = hh;
    cn_out[i * HD + h] = c;
  }
  float s_l = 0.f;
  for (int h = 0; h < HD; ++h) {
    float gi = bih1[h] + bhh1[h], gf = bih1[HD + h] + bhh1[HD + h];
    float gg = bih1[2 * HD + h] + bhh1[2 * HD + h], go = bih1[3 * HD + h] + bhh1[3 * HD + h];
    for (int p = 0; p < HD; ++p) {
      float a = hn0[p], b = h1p[p];
      gi += Wih1[(h)*HD + p] * a + Whh1[(h)*HD + p] * b;
      gf += Wih1[(HD + h) * HD + p] * a + Whh1[(HD + h) * HD + p] * b;
      gg += Wih1[(2 * HD + h) * HD + p] * a + Whh1[(2 * HD + h) * HD + p] * b;
      go += Wih1[(3 * HD + h) * HD + p] * a + Whh1[(3 * HD + h) * HD + p] * b;
    }
    float c = sigf(gf) * c1p[h] + sigf(gi) * tanhf(gg);
    float hh = sigf(go) * tanhf(c);
    hn_out[KH + i * HD + h] = hh;
    cn_out[KH + i * HD + h] = c;
    s_l += Wl[h] * hh;
  }
  lpre[i] = s_l + bl[0];
}

// ============================================================
// 7) mean of lpre (single block, deterministic tree reduce)
// ============================================================
__global__ void k_mean(const float* __restrict__ lpre, float* __restrict__ meanout) {
  __shared__ float sm[256];
  float s = 0.f;
  for (int i = threadIdx.x; i < KD; i += 256) s += lpre[i];
  sm[threadIdx.x] = s; __syncthreads();
  for (int off = 128; off > 0; off >>= 1) {
    if (threadIdx.x < (unsigned)off) sm[threadIdx.x] += sm[threadIdx.x + off];
    __syncthreads();
  }
  if (threadIdx.x == 0) *meanout = sm[0] / (float)KD;
}

// ============================================================
// 8) L = max(diag((lpre-mean)*SCALE + 1), L_before)
// ============================================================
__global__ void k_build_L(const float* __restrict__ lpre, const float* __restrict__ meanp,
                          const float* __restrict__ Lb, float* __restrict__ L) {
  size_t idx = (size_t)blockIdx.x * blockDim.x + threadIdx.x;
  if (idx >= (size_t)KD * KD) return;
  int i = (int)(idx / KD), j = (int)(idx - (size_t)i * KD);
  float dv = 0.f;
  if (i == j) dv = (lpre[i] - *meanp) * SCALE_C + 1.0f;
  L[idx] = fmaxf(dv, Lb[idx]);
}

// ============================================================
// 9) in-place Cholesky of 512x512 Gram (lower), 1 block
// ============================================================
__global__ void k_cholesky(float* __restrict__ g) {
  const int n = N0;
  __shared__ float colj[N0];
  __shared__ float red[N0];
  __shared__ float diag;
  int t = threadIdx.x;
  for (int j = 0; j < n; ++j) {
    for (int p = t; p < j; p += blockDim.x) colj[p] = g[(size_t)j * n + p];
    __syncthreads();
    float s = 0.f;
    for (int p = t; p < j; p += blockDim.x) s += colj[p] * colj[p];
    red[t] = s; __syncthreads();
    for (int off = 256; off > 0; off >>= 1) {
      if (t < off) red[t] += red[t + off];
      __syncthreads();
    }
    if (t == 0) {
      float d = sqrtf(fmaxf(g[(size_t)j * n + j] - red[0], 1e-30f));
      g[(size_t)j * n + j] = d;
      diag = d;
    }
    __syncthreads();
    float d = diag;
    for (int i = j + 1 + t; i < n; i += blockDim.x) {
      const float* gi = g + (size_t)i * n;
      float dot = 0.f;
      for (int p = 0; p < j; ++p) dot += gi[p] * colj[p];
      g[(size_t)i * n + j] = (gi[j] - dot) / d;
    }
    __syncthreads();
  }
}

// ============================================================
// 10) X = C^{-1} (lower-tri), thread per column, upper zeroed
// ============================================================
__global__ void k_trinv(const float* __restrict__ C, float* __restrict__ X) {
  int j = blockIdx.x * blockDim.x + threadIdx.x;
  if (j >= N0) return;
  const int n = N0;
  for (int i = 0; i < j; ++i) X[(size_t)i * n + j] = 0.f;
  X[(size_t)j * n + j] = 1.0f / C[(size_t)j * n + j];
  for (int i = j + 1; i < n; ++i) {
    float s = 0.f;
    for (int p = j; p < i; ++p) s += C[(size_t)i * n + p] * X[(size_t)p * n + j];
    X[(size_t)i * n + j] = -s / C[(size_t)i * n + i];
  }
}

// ============================================================
// host launcher
// ============================================================
extern "C" void kernel_launch(void* const* d_in, const int* in_sizes, int n_in,
                              void* d_out, int out_size, void* d_ws, size_t ws_size,
                              hipStream_t stream) {
  (void)in_sizes; (void)n_in; (void)out_size; (void)ws_size;
  const float* e      = (const float*)d_in[0];
  const float* eg     = (const float*)d_in[1];
  const float* s      = (const float*)d_in[2];   // (512,4608) row-major == M^T
  const float* sg     = (const float*)d_in[3];
  const float* Lh0    = (const float*)d_in[4];
  const float* Lc0    = (const float*)d_in[5];
  const float* Lb     = (const float*)d_in[6];
  const float* Wih0   = (const float*)d_in[7];
  const float* Whh0   = (const float*)d_in[8];
  const float* bih0   = (const float*)d_in[9];
  const float* bhh0   = (const float*)d_in[10];
  const float* Wih1   = (const float*)d_in[11];
  const float* Whh1   = (const float*)d_in[12];
  const float* bih1   = (const float*)d_in[13];
  const float* bhh1   = (const float*)d_in[14];
  const float* Wl     = (const float*)d_in[15];
  const float* bl     = (const float*)d_in[16];
  const float* e_lr   = (const float*)d_in[17];
  const float* s_lr   = (const float*)d_in[18];

  float* out = (float*)d_out;
  const size_t o_e  = 0;
  const size_t o_s  = 16777216;                  // 4096^2
  const size_t o_L  = o_s + 2359296;             // +512*4608
  const size_t o_hn = o_L + (size_t)KD * KD;     // +4608^2
  const size_t o_cn = o_hn + 2ull * KD * HD;

  float* wsf = (float*)d_ws;
  float* A0   = wsf;                             // 512^2
  float* ASYM = wsf + 262144;                    // 512^2
  float* GT   = wsf + 524288;                    // 512*4608
  float* Y0T  = wsf + 2883584;                   // 512*4608
  float* YT   = wsf + 5242880;                   // 512*4608
  float* XV   = wsf + 7602176;                   // 4608
  float* LPRE = wsf + 7606784;                   // 4608
  float* MEAN = wsf + 7611392;                   // 1
  float* GRAM = wsf + 7612416;                   // 512^2
  float* CINV = wsf + 7874560;                   // 512^2

  // 1) euclidean update
  k_e_update<<<(4194304 + 255) / 256, 256, 0, stream>>>(e, eg, e_lr, out + o_e, 4194304);

  // 2) A0 = -s_lr * (S @ Sg^T)   (== M^T Mg)
  k_gemm_abt<<<128, 256, 0, stream>>>(s, sg, A0, N0, N0, KD, -1.0f, s_lr);
  // 3) A_sym
  k_symmetrize<<<1024, 256, 0, stream>>>(A0, ASYM);
  // 4) Gt = -s_lr*Sg - A_sym @ S      (== G^T)
  k_gemm_ab<<<1152, 256, 0, stream>>>(ASYM, s, GT, sg, -1.0f, s_lr, nullptr, 0.0f,
                                      N0, KD, N0, -1.0f);
  // 5) x = clip(log|diag(G G^T)|/p, -1, 1)
  k_coldiag<<<18, 256, 0, stream>>>(GT, XV);
  // 6) LSTM (writes hn/cn outputs)
  k_lstm<<<18, 256, 0, stream>>>(XV, Lh0, Lc0, Wih0, Whh0, bih0, bhh0,
                                 Wih1, Whh1, bih1, bhh1, Wl, bl,
                                 out + o_hn, out + o_cn, LPRE);
  // 7) mean
  k_mean<<<1, 256, 0, stream>>>(LPRE, MEAN);
  // 8) L output
  k_build_L<<<(int)(((size_t)KD * KD + 255) / 256), 256, 0, stream>>>(LPRE, MEAN, Lb, out + o_L);
  // 9) Y0t = Gt @ L^T (== (L @ G)^T) -- 21.7 GFLOP, cooperative LDS-staged kernel
  {
    dim3 grid(KD / 64, N0 / 128);     // 72 x 4 blocks of 256 threads
    k_gemm_abt_coop<<<grid, 256, 0, stream>>>(GT, out + o_L, Y0T, N0, KD, KD, 1.0f);
  }
  // 10) B0 = S @ Y0t^T  (== M^T (L G))
  k_gemm_abt<<<128, 256, 0, stream>>>(s, Y0T, A0, N0, N0, KD, 1.0f, nullptr);
  // 11) B_sym
  k_symmetrize<<<1024, 256, 0, stream>>>(A0, ASYM);
  // 12) Yt = S + Y0t - B_sym @ S   (== (M + P)^T)
  k_gemm_ab<<<1152, 256, 0, stream>>>(ASYM, s, YT, Y0T, 1.0f, nullptr, s, 1.0f,
                                      N0, KD, N0, -1.0f);
  // 13) Gram = Yt @ Yt^T  (== Y^T Y)
  k_gemm_abt<<<128, 256, 0, stream>>>(YT, YT, GRAM, N0, N0, KD, 1.0f, nullptr);
  // 14) Cholesky: Gram -> C (lower), diag > 0 => reference sign-fix is identity
  k_cholesky<<<1, 512, 0, stream>>>(GRAM);
  // 15) Cinv = C^{-1}
  k_trinv<<<2, 256, 0, stream>>>(GRAM, CINV);
  // 16) new_s = Q^T = Cinv @ Yt
  k_gemm_ab<<<1152, 256, 0, stream>>>(CINV, YT, out + o_s, nullptr, 0.0f, nullptr,
                                      nullptr, 0.0f, N0, KD, N0, 1.0f);
}